// MultiResolutionHashEncoder_19945828123192
// MI455X (gfx1250) — compile-verified
//
#include <hip/hip_runtime.h>
#include <stdint.h>

// ---------------- problem constants (from reference) ----------------
#define LEVELS   8
#define FDIM     4
#define LOG2_T   19
#define TBL_ENTRIES (1u << LOG2_T)     // 524288 entries/level, 16B each
#define L_NH     6                     // levels [0,6) use dense (non-hash) indexing
#define PI2      2654435761u
#define PI3      805459861u

// LDS-cached coarse levels: level0 (res 8) max idx = 8+64+512 -> 585 entries,
// level1 (res 16) max idx = 16+256+4096 -> 4369 entries. 4954*16B = 79,264 B LDS
// -> 4 workgroups / WGP (320KB), 8 waves/SIMD.
#define N_LDS_LEVELS 2
#define CAP_L0   585
#define CAP_L1   4369
#define LDS_ENTRIES (CAP_L0 + CAP_L1)

typedef float        fvec4 __attribute__((ext_vector_type(4)));
typedef unsigned int u32x4 __attribute__((ext_vector_type(4)));
typedef int          i32x4 __attribute__((ext_vector_type(4)));
typedef int          i32x8 __attribute__((ext_vector_type(8)));

// LDS-qualified vector type: loads through this MUST lower to ds_load_b128.
typedef __attribute__((address_space(3))) fvec4 lds_fvec4;

#if defined(__has_builtin)
#if __has_builtin(__builtin_amdgcn_tensor_load_to_lds)
#define HAVE_TDM 1
#endif
#endif

__device__ __forceinline__ int imin(int a, int b) { return a < b ? a : b; }

#ifdef HAVE_TDM
// Issue one TDM 1-D copy: n4 4-byte elements from global -> LDS.
// D# built per CDNA5 ISA:
//   group0: count=1 | lds_addr | global_addr[56:0] | type=2
//   group1: data_size=4B, tensor_dim0 = tile_dim0 = n4 (<65536), single row.
__device__ __forceinline__ void tdm_load_1d(unsigned lds_byte_addr,
                                            const void* gptr, unsigned n4) {
  uint64_t ga = (uint64_t)(uintptr_t)gptr;
  u32x4 g0 = { 1u,                                   // count=1 (valid user D#)
               lds_byte_addr,                        // lds_addr (bytes)
               (unsigned)(ga & 0xFFFFFFFFu),         // global_addr[31:0]
               (unsigned)((ga >> 32) & 0x1FFFFFFu)   // global_addr[56:32]
                 | (2u << 30) };                     // type=2 ("image")
  i32x8 g1 = { (int)(2u << 16),                      // data_size=2 -> 4B elems
               (int)((n4 & 0xFFFFu) << 16),          // tensor_dim0[15:0]
               (int)(((n4 >> 16) & 0xFFFFu) | (1u << 16)), // td0[31:16] | tensor_dim1=1
               (int)((n4 & 0xFFFFu) << 16),          // td1[31:16]=0 | tile_dim0=n4
               0,                                    // tile_dim1=0, tile_dim2=0 (unused)
               (int)n4,                              // tensor_dim0_stride[31:0]
               0, 0 };                               // stride hi / dim1 stride = 0
  i32x4 z4 = { 0, 0, 0, 0 };
#if defined(__clang_major__) && (__clang_major__ >= 23)
  i32x8 z8 = { 0, 0, 0, 0, 0, 0, 0, 0 };
  __builtin_amdgcn_tensor_load_to_lds(g0, g1, z4, z4, z8, 0);
#else
  __builtin_amdgcn_tensor_load_to_lds(g0, g1, z4, z4, 0);
#endif
}
#endif

extern "C" __global__ __launch_bounds__(256)
void hashenc_fwd(const float* __restrict__ ev,   // (B*N, 3)
                 const float* __restrict__ hm,   // (8, 2^19, 4)
                 const int*   __restrict__ res,  // (8, 3)
                 float*       __restrict__ out,  // (B*N, 32)
                 int npts) {
  __shared__ fvec4 s_tab[LDS_ENTRIES];

  // 32-bit LDS byte offset of s_tab (flat-address low 32 bits == LDS offset).
  unsigned sbase = (unsigned)(uintptr_t)(void*)&s_tab[0];

  const int lbase[N_LDS_LEVELS] = { 0, CAP_L0 };
  const int lcap [N_LDS_LEVELS] = { CAP_L0, CAP_L1 };
  int  n_ent [N_LDS_LEVELS];
  bool lds_ok[N_LDS_LEVELS];
#pragma unroll
  for (int l = 0; l < N_LDS_LEVELS; ++l) {
    int Rx = res[3*l+0], Ry = res[3*l+1], Rz = res[3*l+2];
    long long mx = 1ll + Rx + (long long)Rx*Ry + (long long)Rx*Ry*Rz;
    n_ent[l]  = (int)mx;
    lds_ok[l] = (mx <= (long long)lcap[l]);
  }

  // ---- stage coarse tables into LDS ----
#ifdef HAVE_TDM
  if (threadIdx.x < 32) {          // wave 0 issues the DMA (EXEC ignored by TDM)
#pragma unroll
    for (int l = 0; l < N_LDS_LEVELS; ++l)
      if (lds_ok[l])
        tdm_load_1d(sbase + (unsigned)lbase[l] * 16u,
                    hm + (size_t)l * TBL_ENTRIES * FDIM,
                    (unsigned)(n_ent[l] * FDIM));
    __builtin_amdgcn_s_wait_tensorcnt(0);
  }
#else
#pragma unroll
  for (int l = 0; l < N_LDS_LEVELS; ++l)
    if (lds_ok[l]) {
      const fvec4* src = (const fvec4*)hm + (size_t)l * TBL_ENTRIES;
      for (int i = threadIdx.x; i < n_ent[l]; i += (int)blockDim.x)
        s_tab[lbase[l] + i] = src[i];
    }
#endif
  // The TDM write to LDS is invisible to the optimizer (the address only
  // escapes as an integer in the D#). Launder the LDS *offset* through a
  // volatile asm (memory clobber), then rebuild an addrspace(3) pointer from
  // it: the loads below cannot be folded away AND must lower to ds_load_b128.
  asm volatile("" : "+s"(sbase) : : "memory");
  lds_fvec4* stab = (lds_fvec4*)(size_t)sbase;
  __syncthreads();

  const int p = (int)(blockIdx.x * blockDim.x + threadIdx.x);
  if (p >= npts) return;

  const float x = ev[3*p+0], y = ev[3*p+1], z = ev[3*p+2];

  fvec4 o[LEVELS];
#pragma unroll
  for (int l = 0; l < LEVELS; ++l) {
    const int Rx = res[3*l+0], Ry = res[3*l+1], Rz = res[3*l+2];
    const float sx = x * (float)Rx, sy = y * (float)Ry, sz = z * (float)Rz;
    const int ix = (int)sx, iy = (int)sy, iz = (int)sz;      // trunc == floor (x>=0)
    const int cx = imin(ix + 1, Rx), cy = imin(iy + 1, Ry), cz = imin(iz + 1, Rz);
    const float wx0 = 1.f - fabsf((float)ix - sx), wx1 = 1.f - fabsf((float)cx - sx);
    const float wy0 = 1.f - fabsf((float)iy - sy), wy1 = 1.f - fabsf((float)cy - sy);
    const float wz0 = 1.f - fabsf((float)iz - sz), wz1 = 1.f - fabsf((float)cz - sz);
    const fvec4* __restrict__ gtab = (const fvec4*)hm + (size_t)l * TBL_ENTRIES;

    fvec4 acc = { 0.f, 0.f, 0.f, 0.f };
#pragma unroll
    for (int k = 0; k < 8; ++k) {                 // corner combo: bit2->x, bit1->y, bit0->z
      const int gx = (k & 4) ? cx : ix;
      const int gy = (k & 2) ? cy : iy;
      const int gz = (k & 1) ? cz : iz;
      const float w = ((k & 4) ? wx1 : wx0) * ((k & 2) ? wy1 : wy0) * ((k & 1) ? wz1 : wz0);
      unsigned idx;
      if (l < L_NH)
        idx = (unsigned)(gx + gy * Rx + gz * (Rx * Ry));
      else
        idx = ((unsigned)gx ^ (unsigned)gy * PI2 ^ (unsigned)gz * PI3) & (TBL_ENTRIES - 1u);
      fvec4 f;
      if (l < N_LDS_LEVELS && lds_ok[l]) f = stab[lbase[l] + (int)idx];   // ds_load_b128
      else                               f = gtab[idx];                   // global_load_b128
      acc += f * w;                                                       // 4x v_fma_f32
    }
    o[l] = acc;
  }

  fvec4* __restrict__ op = (fvec4*)out + (size_t)p * LEVELS;  // 128B contiguous per point
#pragma unroll
  for (int l = 0; l < LEVELS; ++l) op[l] = o[l];               // 8x global_store_b128
}

extern "C" void kernel_launch(void* const* d_in, const int* in_sizes, int n_in,
                              void* d_out, int out_size, void* d_ws, size_t ws_size,
                              hipStream_t stream) {
  const float* ev  = (const float*)d_in[0];
  const float* hm  = (const float*)d_in[1];
  const int*   res = (const int*)d_in[2];
  float*       out = (float*)d_out;
  const int npts  = in_sizes[0] / 3;        // (B*N) points
  const int block = 256;                     // 8 wave32 waves
  const int grid  = (npts + block - 1) / block;
  hashenc_fwd<<<grid, block, 0, stream>>>(ev, hm, res, out, npts);
}